// AttentionHead_45664092291306
// MI455X (gfx1250) — compile-verified
//
#include <hip/hip_runtime.h>
#include <hip/hip_bf16.h>
#include <math.h>

// ---------------------------------------------------------------------------
// GAT attention head for MI455X (gfx1250, wave32).
// Bandwidth-bound workload; GEMM done with V_WMMA_F32_16X16X4_F32 (full fp32).
// ---------------------------------------------------------------------------

typedef float v2f __attribute__((ext_vector_type(2)));
typedef float v8f __attribute__((ext_vector_type(8)));

#define GAT_IN_DIM 256
#define GAT_OUT_DIM 32
#define GAT_K2 (GAT_IN_DIM / 2)   // 128 K-pairs
#define GAT_WSTRIDE 33            // padded n-stride (conflict-avoiding)

// ---------------------------------------------------------------------------
// K0: init accumulation state (d_out zeroed; max=-inf; sum=0)
// ---------------------------------------------------------------------------
__global__ void gat_init_kernel(float* __restrict__ out_accum,
                                float* __restrict__ maxb,
                                float* __restrict__ sumb, int N) {
  int i = blockIdx.x * blockDim.x + threadIdx.x;
  int total = N * GAT_OUT_DIM;
  if (i < total) out_accum[i] = 0.0f;
  if (i < N) {
    maxb[i] = -INFINITY;
    sumb[i] = 0.0f;
  }
}

// ---------------------------------------------------------------------------
// K1: h_prime[N,32] = X[N,256] @ W[32,256]^T  via V_WMMA_F32_16X16X4_F32.
// One wave computes a 16x32 output tile (two 16x16 accumulators).
//
// W is staged into LDS as (K,K+1) float2 pairs: Wp[k2 * 33 + n] = {W[n][2k2],
// W[n][2k2+1]}. A B fragment is then ONE ds_load_b64 directly into the
// even-aligned VGPR pair WMMA wants (no cross-register fixup movs), and
// consecutive lanes land 8B apart -> conflict-light.
//
// Fragment layouts (ISA 7.12.2, wave32):
//   A (16x4 f32, 2 VGPRs): lane L -> M = L%16 ; VGPR v holds K = v + 2*(L/16)
//   B (4x16 f32, 2 VGPRs): lane L -> N = L%16 ; VGPR v holds K = v + 2*(L/16)
//   C (16x16 f32, 8 VGPRs): VGPR r -> M = r + 8*(L/16), N = L%16
// ---------------------------------------------------------------------------
__global__ void __launch_bounds__(256) gat_gemm_kernel(
    const float* __restrict__ X,  // [N, 256]
    const float* __restrict__ W,  // [32, 256]
    float* __restrict__ H,        // [N, 32]
    int N) {
  __shared__ v2f Wp[GAT_K2 * GAT_WSTRIDE];  // 128 x 33 float2 = 33 KB

  for (int i = threadIdx.x; i < GAT_OUT_DIM * GAT_K2; i += blockDim.x) {
    int o = i / GAT_K2;   // output channel n
    int k2 = i % GAT_K2;  // K-pair index
    Wp[k2 * GAT_WSTRIDE + o] = *(const v2f*)(W + (size_t)o * GAT_IN_DIM + 2 * k2);
  }
  __syncthreads();

  const int lane = threadIdx.x & 31;
  const int wave = threadIdx.x >> 5;
  const int rowBase = (blockIdx.x * 8 + wave) * 16;
  if (rowBase >= N) return;  // uniform per-wave: EXEC stays all-ones inside

  const int mrow = lane & 15;  // M (for A) and N (for B/C) sub-index
  const int hi = lane >> 4;    // lane half: selects K+2 (A/B) / M+8 (C)

  int row = rowBase + mrow;
  if (row >= N) row = N - 1;  // clamp loads; stores are guarded below
  const float* xrow = X + (size_t)row * GAT_IN_DIM;

  // Pull the back half of this lane's row toward the WGP cache early
  // (global_prefetch_b8); the front half is consumed immediately anyway.
  __builtin_prefetch(xrow + GAT_IN_DIM / 2, 0, 0);

  v8f acc0 = {};
  v8f acc1 = {};

#pragma unroll 4
  for (int t = 0; t < GAT_IN_DIM / 4; ++t) {
    const int k2 = 2 * t + hi;        // this lane-half's K-pair
    v2f a = *(const v2f*)(xrow + 2 * k2);            // A: K = 2k2, 2k2+1
    v2f b0 = Wp[k2 * GAT_WSTRIDE + mrow];            // B, n-tile 0
    v2f b1 = Wp[k2 * GAT_WSTRIDE + 16 + mrow];       // B, n-tile 1
    acc0 = __builtin_amdgcn_wmma_f32_16x16x4_f32(false, a, false, b0,
                                                 (short)0, acc0, false, false);
    acc1 = __builtin_amdgcn_wmma_f32_16x16x4_f32(false, a, false, b1,
                                                 (short)0, acc1, false, false);
  }

  if (rowBase + 16 <= N) {
    // Fast path (all but at most the last wave): no per-element guards.
#pragma unroll
    for (int r = 0; r < 8; ++r) {
      int m = rowBase + 8 * hi + r;
      H[(size_t)m * GAT_OUT_DIM + mrow] = acc0[r];
      H[(size_t)m * GAT_OUT_DIM + 16 + mrow] = acc1[r];
    }
  } else {
#pragma unroll
    for (int r = 0; r < 8; ++r) {
      int m = rowBase + 8 * hi + r;
      if (m < N) {
        H[(size_t)m * GAT_OUT_DIM + mrow] = acc0[r];
        H[(size_t)m * GAT_OUT_DIM + 16 + mrow] = acc1[r];
      }
    }
  }
}

// ---------------------------------------------------------------------------
// K2: per-node attention logits. One wave32 per node; lane = channel.
// e_dst[n] = h'[n]·a_dst_w + a_dst_b ; e_src[n] = h'[n]·a_src_w + a_src_b
// ---------------------------------------------------------------------------
__global__ void __launch_bounds__(256) gat_logits_kernel(
    const float* __restrict__ H, const float* __restrict__ adw,
    const float* __restrict__ adb, const float* __restrict__ asw,
    const float* __restrict__ asb, float* __restrict__ e_dst,
    float* __restrict__ e_src, int N) {
  int node = (blockIdx.x * blockDim.x + threadIdx.x) >> 5;
  int lane = threadIdx.x & 31;
  if (node >= N) return;
  float h = H[(size_t)node * GAT_OUT_DIM + lane];
  float pd = h * adw[lane];
  float ps = h * asw[lane];
#pragma unroll
  for (int off = 16; off > 0; off >>= 1) {
    pd += __shfl_xor(pd, off, 32);
    ps += __shfl_xor(ps, off, 32);
  }
  if (lane == 0) {
    e_dst[node] = pd + adb[0];
    e_src[node] = ps + asb[0];
  }
}

// ---------------------------------------------------------------------------
// K3: per-edge leaky-relu score + segment max via monotone float atomic trick.
// ---------------------------------------------------------------------------
__device__ __forceinline__ void atomic_max_float(float* addr, float v) {
  if (v >= 0.0f)
    atomicMax((int*)addr, __float_as_int(v));
  else
    atomicMin((unsigned int*)addr, __float_as_uint(v));
}

__global__ void gat_scores_kernel(const int* __restrict__ esrc,
                                  const int* __restrict__ edst,
                                  const float* __restrict__ e_dst,
                                  const float* __restrict__ e_src,
                                  float* __restrict__ scores,
                                  float* __restrict__ maxb, int E) {
  int e = blockIdx.x * blockDim.x + threadIdx.x;
  if (e >= E) return;
  int d = edst[e];
  int s = esrc[e];
  float x = e_dst[d] + e_src[s];
  x = (x > 0.0f) ? x : 0.2f * x;  // leaky relu, slope 0.2
  scores[e] = x;
  atomic_max_float(maxb + d, x);
}

// ---------------------------------------------------------------------------
// K4: per-edge exp(score - max[dst]) (stored in place) + segment sum.
// ---------------------------------------------------------------------------
__global__ void gat_exp_kernel(const int* __restrict__ edst,
                               const float* __restrict__ maxb,
                               float* __restrict__ scores,
                               float* __restrict__ sumb, int E) {
  int e = blockIdx.x * blockDim.x + threadIdx.x;
  if (e >= E) return;
  int d = edst[e];
  float ex = __expf(scores[e] - maxb[d]);
  scores[e] = ex;  // reuse slot: scores -> exp_scores
  atomicAdd(sumb + d, ex);
}

// ---------------------------------------------------------------------------
// K5: per-node reciprocal of segment sum (in place).
// ---------------------------------------------------------------------------
__global__ void gat_inv_kernel(float* __restrict__ sumb, int N) {
  int i = blockIdx.x * blockDim.x + threadIdx.x;
  if (i < N) sumb[i] = 1.0f / fmaxf(sumb[i], 1e-12f);
}

// ---------------------------------------------------------------------------
// K6: weighted aggregation. One wave32 per edge, lane = output channel:
// out[dst*32+lane] += h'[src*32+lane] * (exp[e] * inv[dst]).
// 128B coalesced gather + 128B coalesced atomic burst per edge (L2-resident).
// ---------------------------------------------------------------------------
__global__ void __launch_bounds__(256) gat_aggregate_kernel(
    const int* __restrict__ esrc, const int* __restrict__ edst,
    const float* __restrict__ H, const float* __restrict__ exps,
    const float* __restrict__ inv, float* __restrict__ out_accum, int E) {
  int e = (blockIdx.x * blockDim.x + threadIdx.x) >> 5;
  int lane = threadIdx.x & 31;
  if (e >= E) return;
  int s = esrc[e];
  int d = edst[e];
  float alpha = exps[e] * inv[d];
  float v = H[(size_t)s * GAT_OUT_DIM + lane] * alpha;
  atomicAdd(out_accum + (size_t)d * GAT_OUT_DIM + lane, v);
}

// ---------------------------------------------------------------------------
// K7: ELU in place on d_out.
// ---------------------------------------------------------------------------
__global__ void gat_elu_kernel(float* __restrict__ out, int total) {
  int i = blockIdx.x * blockDim.x + threadIdx.x;
  if (i < total) {
    float x = out[i];
    out[i] = (x > 0.0f) ? x : (__expf(x) - 1.0f);
  }
}

// ---------------------------------------------------------------------------
// Host launcher
// ---------------------------------------------------------------------------
extern "C" void kernel_launch(void* const* d_in, const int* in_sizes, int n_in,
                              void* d_out, int out_size, void* d_ws,
                              size_t ws_size, hipStream_t stream) {
  const float* X = (const float*)d_in[0];    // [N, 256]
  const int* esrc = (const int*)d_in[1];     // [E]
  const int* edst = (const int*)d_in[2];     // [E]
  const float* W = (const float*)d_in[4];    // [32, 256]
  const float* adw = (const float*)d_in[5];  // [32]
  const float* adb = (const float*)d_in[6];  // scalar
  const float* asw = (const float*)d_in[7];  // [32]
  const float* asb = (const float*)d_in[8];  // scalar

  const int N = in_sizes[0] / GAT_IN_DIM;
  const int E = in_sizes[1];

  // Workspace (floats): H[N*32] | e_dst[N] | e_src[N] | max[N] | sum[N] | scores[E]
  float* ws = (float*)d_ws;
  float* H = ws;
  float* e_dst = H + (size_t)N * GAT_OUT_DIM;
  float* e_src = e_dst + N;
  float* maxb = e_src + N;
  float* sumb = maxb + N;
  float* scores = sumb + N;

  float* out = (float*)d_out;  // [N, 32], accumulated in place then ELU'd

  const int BT = 256;

  // K0: init
  {
    int total = N * GAT_OUT_DIM;
    gat_init_kernel<<<(total + BT - 1) / BT, BT, 0, stream>>>(out, maxb, sumb,
                                                              N);
  }
  // K1: WMMA GEMM
  {
    int nWaves = (N + 15) / 16;
    int blocks = (nWaves + 7) / 8;  // 8 waves (256 threads) per block
    gat_gemm_kernel<<<blocks, BT, 0, stream>>>(X, W, H, N);
  }
  // K2: node logits
  {
    int blocks = (N + 7) / 8;  // 8 nodes per 256-thread block
    gat_logits_kernel<<<blocks, BT, 0, stream>>>(H, adw, adb, asw, asb, e_dst,
                                                 e_src, N);
  }
  // K3: edge scores + segment max
  gat_scores_kernel<<<(E + BT - 1) / BT, BT, 0, stream>>>(esrc, edst, e_dst,
                                                          e_src, scores, maxb,
                                                          E);
  // K4: exp + segment sum
  gat_exp_kernel<<<(E + BT - 1) / BT, BT, 0, stream>>>(edst, maxb, scores,
                                                       sumb, E);
  // K5: reciprocal
  gat_inv_kernel<<<(N + BT - 1) / BT, BT, 0, stream>>>(sumb, N);
  // K6: aggregation (wave per edge)
  {
    int blocks = (E + 7) / 8;  // 8 edges (waves) per 256-thread block
    gat_aggregate_kernel<<<blocks, BT, 0, stream>>>(esrc, edst, H, scores,
                                                    sumb, out, E);
  }
  // K7: ELU
  {
    int total = N * GAT_OUT_DIM;
    gat_elu_kernel<<<(total + BT - 1) / BT, BT, 0, stream>>>(out, total);
  }
}